// AssociativeAttention_14104672600717
// MI455X (gfx1250) — compile-verified
//
#include <hip/hip_runtime.h>
#include <math.h>

// AssociativeAttention for MI455X (gfx1250, wave32, WMMA).
// All matmul stages use V_WMMA_F32_16X16X4_F32 (full fp32 precision; the
// workload is bandwidth-bound at 23.3 TB/s so f32 WMMA peak is not the
// limiter). Operand panels are staged into LDS with the gfx1250 async
// global->LDS path (ASYNCcnt), double-buffered so WMMA overlaps the copies.

typedef float v2f __attribute__((ext_vector_type(2)));
typedef float v8f __attribute__((ext_vector_type(8)));

#define WMMA_F32(a, b, c) \
    __builtin_amdgcn_wmma_f32_16x16x4_f32(false, (a), false, (b), (short)0, (c), false, false)

// ---- gfx1250 async global->LDS copy (ASYNCcnt) with safe fallback ---------
#if __has_builtin(__builtin_amdgcn_global_load_async_to_lds_b128)
#define HAVE_ASYNC_LDS 1
typedef int v4i_t __attribute__((ext_vector_type(4)));
typedef __attribute__((address_space(1))) v4i_t* gv4i_p;
typedef __attribute__((address_space(3))) v4i_t* lv4i_p;
#define ASYNC_CP16(gsrc, ldst)                                              \
    __builtin_amdgcn_global_load_async_to_lds_b128(                         \
        (gv4i_p)(void*)(gsrc), (lv4i_p)(void*)(ldst), 0, 0)
#else
#define HAVE_ASYNC_LDS 0
#define ASYNC_CP16(gsrc, ldst)                                              \
    do { *(float4*)(ldst) = *(const float4*)(gsrc); } while (0)
#endif

#if __has_builtin(__builtin_amdgcn_s_wait_asynccnt)
#define WAIT_ASYNC(n) __builtin_amdgcn_s_wait_asynccnt(n)   // n must be a literal
#elif HAVE_ASYNC_LDS
#define WAIT_ASYNC(n) asm volatile("s_wait_asynccnt %0" ::"i"(n) : "memory")
#else
#define WAIT_ASYNC(n) ((void)0)
#endif

static constexpr int Lq = 1024, Dd = 512, Hh = 8, Kf = 24;
static constexpr float EPSF = 1e-5f;

// ---------------------------------------------------------------------------
// f32 WMMA GEMM, compile-time dims: C[MxN] = A[MxK] @ B[KxN] + bias[N].
// Block = 128 threads = 4 waves; tile = 64(m) x 64(n); wave w owns rows
// [bm*64+16w, +16) x all 64 cols (4 accumulators, A frag reused 4x).
// B panel (16k x 64n) is async-staged into double-buffered LDS.
// OUTMODE 0: row-major [MxN]; OUTMODE 1: head-split [n>>6][m][n&63].
// ---------------------------------------------------------------------------
template <int M, int N, int K, int OUTMODE>
__global__ void gemm_f32_wmma(const float* __restrict__ A,
                              const float* __restrict__ B,
                              const float* __restrict__ bias,
                              float* __restrict__ C)
{
    constexpr int NT = N / 64;
    constexpr int NCH = K / 16;                       // k-chunks of 16
    const int bm = blockIdx.x / NT, bn = blockIdx.x % NT;
    const int lane = threadIdx.x & 31;
    const int wave = threadIdx.x >> 5;
    const int half = lane >> 4, lm = lane & 15;
    const int m0 = bm * 64 + wave * 16;
    const int n0 = bn * 64;

    __shared__ __align__(16) float bs[2][16 * 64];    // 8KB double-buffered B

    const int li = threadIdx.x * 8;                   // 8 floats per thread
    const int br = li >> 6, bc = li & 63;

    // stage B chunk kc into buffer buf (2 async b128 per thread per chunk)
    auto issueB = [&](int kc, int buf) {
        const float* src = B + (size_t)(kc * 16 + br) * N + n0 + bc;
        ASYNC_CP16(src,     &bs[buf][li]);
        ASYNC_CP16(src + 4, &bs[buf][li + 4]);
    };

    const float* arow = A + (size_t)(m0 + lm) * K;
    v8f acc[4] = {{}, {}, {}, {}};

    issueB(0, 0);
    for (int kc = 0; kc < NCH; ++kc) {
        const int buf = kc & 1;
        if (kc + 1 < NCH) {
            issueB(kc + 1, buf ^ 1);
            WAIT_ASYNC(2);                            // chunk kc resident
        } else {
            WAIT_ASYNC(0);
        }
        __syncthreads();
        __builtin_prefetch(arow + (kc + 1) * 16, 0, 1);
        for (int ks = 0; ks < 4; ++ks) {
            const int kk = ks * 4 + 2 * half;
            v2f a;
            a.x = arow[kc * 16 + kk];
            a.y = arow[kc * 16 + kk + 1];
            for (int eb = 0; eb < 4; ++eb) {
                v2f b;
                b.x = bs[buf][kk * 64 + eb * 16 + lm];
                b.y = bs[buf][(kk + 1) * 64 + eb * 16 + lm];
                acc[eb] = WMMA_F32(a, b, acc[eb]);
            }
        }
        __syncthreads();                              // before buf reuse
    }

    for (int eb = 0; eb < 4; ++eb)
        for (int j = 0; j < 8; ++j) {
            const int m = m0 + j + 8 * half;
            const int n = n0 + eb * 16 + lm;
            const float v = acc[eb][j] + bias[n];
            if (OUTMODE == 0) C[(size_t)m * N + n] = v;
            else              C[(size_t)(n >> 6) * (M * 64) + m * 64 + (n & 63)] = v;
        }
}

// ---------------------------------------------------------------------------
// sim = qk_scale[h] * dot(q,k); then L2-normalize k,v rows in place.
// ---------------------------------------------------------------------------
__global__ void simnorm_kernel(const float* __restrict__ qh,
                               float* __restrict__ kh,
                               float* __restrict__ vh,
                               const float* __restrict__ qkscale,
                               float* __restrict__ sim)
{
    const int lane = threadIdx.x & 31;
    const int wave = threadIdx.x >> 5;
    const int site = blockIdx.x * (blockDim.x >> 5) + wave;   // h*1024 + l
    const int h    = site >> 10;
    const size_t base = (size_t)site * 64 + 2 * lane;

    const float q0 = qh[base], q1 = qh[base + 1];
    const float k0 = kh[base], k1 = kh[base + 1];
    const float v0 = vh[base], v1 = vh[base + 1];

    float dqk = q0 * k0 + q1 * k1;
    float nk  = k0 * k0 + k1 * k1;
    float nv  = v0 * v0 + v1 * v1;
    for (int off = 16; off; off >>= 1) {
        dqk += __shfl_xor(dqk, off);
        nk  += __shfl_xor(nk,  off);
        nv  += __shfl_xor(nv,  off);
    }
    const float rk = 1.0f / fmaxf(sqrtf(nk), 1e-12f);
    const float rv = 1.0f / fmaxf(sqrtf(nv), 1e-12f);
    kh[base] = k0 * rk; kh[base + 1] = k1 * rk;
    vh[base] = v0 * rv; vh[base + 1] = v1 * rv;
    if (lane == 0) sim[site] = qkscale[h] * dqk;
}

// ---------------------------------------------------------------------------
// Causal Toeplitz "FFT conv" via WMMA:
//   Y[l,kf,d] = sum_{t<=l} filters[l-t,kf] * u[t,d]
// Grid: (L/16, Kf, H); block 128 = 4 waves (wave = d-block of 16).
// Filter column in LDS; k/v activation chunks async double-buffered in LDS;
// one triangular A fragment feeds both the k-conv and v-conv wmma.
// ---------------------------------------------------------------------------
__global__ void conv_wmma_kernel(const float* __restrict__ kh,
                                 const float* __restrict__ vh,
                                 const float* __restrict__ filt,
                                 float* __restrict__ Kc,
                                 float* __restrict__ Vc)
{
    __shared__ __align__(16) float fcol[Lq];
    __shared__ __align__(16) float kbuf[2][16 * 64];
    __shared__ __align__(16) float vbuf[2][16 * 64];

    const int kf   = blockIdx.y;
    const int head = blockIdx.z;
    const int l0   = blockIdx.x << 4;
    for (int i = threadIdx.x; i < Lq; i += blockDim.x) fcol[i] = filt[i * Kf + kf];

    const int lane = threadIdx.x & 31;
    const int wave = threadIdx.x >> 5;
    const int d0   = wave << 4;
    const int half = lane >> 4, lm = lane & 15;
    const int l    = l0 + lm;

    const float* ku = kh + (size_t)head * Lq * 64;
    const float* vu = vh + (size_t)head * Lq * 64;

    const int li = threadIdx.x * 8;
    const int br = li >> 6, bc = li & 63;
    auto issueU = [&](int kc, int buf) {
        const float* ks = ku + (size_t)(kc * 16 + br) * 64 + bc;
        const float* vs = vu + (size_t)(kc * 16 + br) * 64 + bc;
        ASYNC_CP16(ks,     &kbuf[buf][li]);
        ASYNC_CP16(ks + 4, &kbuf[buf][li + 4]);
        ASYNC_CP16(vs,     &vbuf[buf][li]);
        ASYNC_CP16(vs + 4, &vbuf[buf][li + 4]);
    };

    const int nc = (l0 >> 4) + 1;                      // chunks of 16 t's
    v8f ak = {}, av = {};
    issueU(0, 0);
    for (int kc = 0; kc < nc; ++kc) {
        const int buf = kc & 1;
        if (kc + 1 < nc) {
            issueU(kc + 1, buf ^ 1);
            WAIT_ASYNC(4);
        } else {
            WAIT_ASYNC(0);
        }
        __syncthreads();                               // also covers fcol
        for (int ks = 0; ks < 4; ++ks) {
            const int rr = ks * 4 + 2 * half;
            const int tt = kc * 16 + rr;
            v2f a, bk, bv;
            a.x = (tt     <= l) ? fcol[l - tt]     : 0.0f;
            a.y = (tt + 1 <= l) ? fcol[l - tt - 1] : 0.0f;
            bk.x = kbuf[buf][rr * 64 + d0 + lm];
            bk.y = kbuf[buf][(rr + 1) * 64 + d0 + lm];
            bv.x = vbuf[buf][rr * 64 + d0 + lm];
            bv.y = vbuf[buf][(rr + 1) * 64 + d0 + lm];
            ak = WMMA_F32(a, bk, ak);
            av = WMMA_F32(a, bv, av);
        }
        __syncthreads();
    }
    for (int j = 0; j < 8; ++j) {
        const int ll = l0 + j + 8 * half;
        const size_t idx = (((size_t)head * Lq + ll) * Kf + kf) * 64 + d0 + lm;
        Kc[idx] = ak[j];
        Vc[idx] = av[j];
    }
}

// ---------------------------------------------------------------------------
// Per (h,l): Z[d,e] = sum_kf Vc[l,kf,d]*Kc[l,kf,e]  (WMMA over the 24 kf)
// then Z *= kv_norm_scale[h,d,e]; gate = relu(Z.Wg + bg)^2 + eps;
// write gated_Z and gate. Vc/Kc panels async-staged to LDS (shared by waves).
// ---------------------------------------------------------------------------
__global__ void z_gate_kernel(const float* __restrict__ Kc,
                              const float* __restrict__ Vc,
                              const float* __restrict__ Wg,
                              const float* __restrict__ bg,
                              const float* __restrict__ kvs,
                              float* __restrict__ gZ,
                              float* __restrict__ gts)
{
    __shared__ __align__(16) float vls[Kf * 64];       // 6KB
    __shared__ __align__(16) float kls[Kf * 64];       // 6KB
    __shared__ float red;

    const int site = blockIdx.x;                       // h*1024 + l
    const int h    = site >> 10;
    const int lane = threadIdx.x & 31;
    const int wave = threadIdx.x >> 5;
    const int half = lane >> 4, lm = lane & 15;
    const int m0   = wave << 4;

    const float* Vl = Vc + (size_t)site * Kf * 64;
    const float* Kl = Kc + (size_t)site * Kf * 64;
    for (int i = 0; i < 3; ++i) {                      // 12 floats per thread each
        const int li = threadIdx.x * 12 + i * 4;
        ASYNC_CP16(Vl + li, &vls[li]);
        ASYNC_CP16(Kl + li, &kls[li]);
    }
    if (threadIdx.x == 0) red = 0.0f;
    WAIT_ASYNC(0);
    __syncthreads();

    v8f acc[4] = {{}, {}, {}, {}};
    for (int kk0 = 0; kk0 < Kf; kk0 += 4) {
        const int kk = kk0 + 2 * half;
        v2f a;
        a.x = vls[kk * 64 + m0 + lm];                  // A[m=d, k=kf] = Vc^T
        a.y = vls[(kk + 1) * 64 + m0 + lm];
        for (int eb = 0; eb < 4; ++eb) {
            v2f b;
            b.x = kls[kk * 64 + eb * 16 + lm];
            b.y = kls[(kk + 1) * 64 + eb * 16 + lm];
            acc[eb] = WMMA_F32(a, b, acc[eb]);
        }
    }

    float part = 0.0f;
    for (int eb = 0; eb < 4; ++eb)
        for (int j = 0; j < 8; ++j) {
            const int m = m0 + j + 8 * half;
            const int n = eb * 16 + lm;
            const float z = acc[eb][j] * kvs[h * 4096 + m * 64 + n];
            acc[eb][j] = z;
            part += z * Wg[m * 64 + n];
        }
    for (int off = 16; off; off >>= 1) part += __shfl_xor(part, off);
    if (lane == 0) atomicAdd(&red, part);
    __syncthreads();

    const float r    = fmaxf(red + bg[0], 0.0f);
    const float gate = r * r + EPSF;

    float* out = gZ + (size_t)site * 4096;
    for (int eb = 0; eb < 4; ++eb)
        for (int j = 0; j < 8; ++j) {
            const int m = m0 + j + 8 * half;
            const int n = eb * 16 + lm;
            out[m * 64 + n] = gate * acc[eb][j];
        }
    if (threadIdx.x == 0) gts[site] = gate;
}

// ---------------------------------------------------------------------------
// In-place cumsum of gated_Z over l. Thread per (h, d*64+e) -> coalesced.
// ---------------------------------------------------------------------------
__global__ void cumsum_gz_kernel(float* __restrict__ gZ)
{
    const int tid = blockIdx.x * blockDim.x + threadIdx.x;   // 0..32767
    const int h   = tid >> 12;
    const int de  = tid & 4095;
    size_t idx = (size_t)h * Lq * 4096 + de;
    float acc = 0.0f;
    for (int l = 0; l < Lq; ++l, idx += 4096) {
        acc += gZ[idx];
        gZ[idx] = acc;
    }
}

// ---------------------------------------------------------------------------
// Online-softmax (m,s) scan + gate cumsum, sequential over l per head (tiny).
// ---------------------------------------------------------------------------
__global__ void scans_kernel(const float* __restrict__ sim,
                             const float* __restrict__ gts,
                             float* __restrict__ msc,
                             float* __restrict__ ssc,
                             float* __restrict__ gcum)
{
    const int h = threadIdx.x;
    if (h >= Hh) return;
    float m = -__builtin_inff(), s = 0.0f, g = 0.0f;
    for (int l = 0; l < Lq; ++l) {
        const int i = h * Lq + l;
        const float x  = sim[i];
        const float mn = fmaxf(m, x);
        s = s * expf(m - mn) + expf(x - mn);
        m = mn;
        g += gts[i];
        msc[i] = m; ssc[i] = s; gcum[i] = g;
    }
}

// ---------------------------------------------------------------------------
// ctxt[h,l,e] = scal * sum_d q[h,l,d]*Zc[h,l,d,e]
//   scal = (1 + silu(w)) / (gcum + eps), w = exp(sim-m)/(s+eps) (scalar/site)
// ---------------------------------------------------------------------------
__global__ void ctxt_kernel(const float* __restrict__ qh,
                            const float* __restrict__ Zc,
                            const float* __restrict__ gcum,
                            const float* __restrict__ sim,
                            const float* __restrict__ msc,
                            const float* __restrict__ ssc,
                            float* __restrict__ ct)
{
    const int h    = blockIdx.y;
    const int l    = blockIdx.x * 4 + (threadIdx.x >> 6);
    const int e    = threadIdx.x & 63;
    const int site = h * Lq + l;

    const float w    = expf(sim[site] - msc[site]) / (ssc[site] + EPSF);
    const float silu = w / (1.0f + expf(-w));
    const float scal = (1.0f + silu) / (gcum[site] + EPSF);

    const float* q = qh + (size_t)site * 64;
    const float* Z = Zc + (size_t)site * 4096;
    float acc = 0.0f;
    for (int d = 0; d < 64; ++d) acc += q[d] * Z[d * 64 + e];
    ct[(size_t)l * Dd + h * 64 + e] = acc * scal;
}

// ---------------------------------------------------------------------------
extern "C" void kernel_launch(void* const* d_in, const int* in_sizes, int n_in,
                              void* d_out, int out_size, void* d_ws, size_t ws_size,
                              hipStream_t stream)
{
    const float* x    = (const float*)d_in[0];
    const float* Wq   = (const float*)d_in[1];
    const float* bq   = (const float*)d_in[2];
    const float* Wk   = (const float*)d_in[3];
    const float* bk   = (const float*)d_in[4];
    const float* Wv   = (const float*)d_in[5];
    const float* bv   = (const float*)d_in[6];
    const float* Wo   = (const float*)d_in[7];
    const float* bo   = (const float*)d_in[8];
    const float* Wg   = (const float*)d_in[9];
    const float* bg   = (const float*)d_in[10];
    const float* kvs  = (const float*)d_in[11];   // kv_norm_scale [H,64,64]
    const float* qks  = (const float*)d_in[12];   // qk_norm_scale [H]
    const float* filt = (const float*)d_in[13];   // spectral_filters [L,K]
    float* out = (float*)d_out;

    float* w = (float*)d_ws;
    float* qh   = w; w += (size_t)Hh * Lq * 64;
    float* kh   = w; w += (size_t)Hh * Lq * 64;
    float* vh   = w; w += (size_t)Hh * Lq * 64;
    float* sim  = w; w += Hh * Lq;
    float* msc  = w; w += Hh * Lq;
    float* ssc  = w; w += Hh * Lq;
    float* gts  = w; w += Hh * Lq;
    float* gcum = w; w += Hh * Lq;
    float* ct   = w; w += (size_t)Lq * Dd;
    float* Kc   = w; w += (size_t)Hh * Lq * Kf * 64;
    float* Vc   = w; w += (size_t)Hh * Lq * Kf * 64;
    float* gZ   = w;  // [H][L][4096]; cumsum'd in place -> Z_scan

    // 1) q/k/v projections (WMMA GEMM, async-staged B panels, head-split out)
    {
        dim3 grid((Lq / 64) * (Dd / 64)), block(128);     // 128 blocks
        gemm_f32_wmma<Lq, Dd, Dd, 1><<<grid, block, 0, stream>>>(x, Wq, bq, qh);
        gemm_f32_wmma<Lq, Dd, Dd, 1><<<grid, block, 0, stream>>>(x, Wk, bk, kh);
        gemm_f32_wmma<Lq, Dd, Dd, 1><<<grid, block, 0, stream>>>(x, Wv, bv, vh);
    }
    // 2) sim + in-place normalize k,v
    simnorm_kernel<<<dim3(Hh * Lq / 8), dim3(256), 0, stream>>>(qh, kh, vh, qks, sim);
    // 3) causal Toeplitz conv (WMMA + async double-buffered LDS)
    conv_wmma_kernel<<<dim3(Lq / 16, Kf, Hh), dim3(128), 0, stream>>>(kh, vh, filt, Kc, Vc);
    // 4) Z einsum (WMMA, LDS-staged panels) + kv scale + gate
    z_gate_kernel<<<dim3(Hh * Lq), dim3(128), 0, stream>>>(Kc, Vc, Wg, bg, kvs, gZ, gts);
    // 5) cumsum of gated_Z over l (in place); (m,s) + gate scans
    cumsum_gz_kernel<<<dim3(128), dim3(256), 0, stream>>>(gZ);
    scans_kernel<<<dim3(1), dim3(32), 0, stream>>>(sim, gts, msc, ssc, gcum);
    // 6) ctxt matvec per site
    ctxt_kernel<<<dim3(Lq / 4, Hh), dim3(256), 0, stream>>>(qh, gZ, gcum, sim, msc, ssc, ct);
    // 7) output projection (WMMA GEMM, row-major out)
    gemm_f32_wmma<Lq, Dd, Dd, 0><<<dim3((Lq / 64) * (Dd / 64)), dim3(128), 0, stream>>>(ct, Wo, bo, out);
}